// Lidar2BEV_34591666602814
// MI455X (gfx1250) — compile-verified
//
#include <hip/hip_runtime.h>
#include <hip/hip_bf16.h>

// BEV voxelization: zero a dense [B,512,512,40,3] f32 grid, then scatter-add
// point xyz into voxels via f32 global atomics (no-return -> resolve in L2).
// Memory-bound: 251.7 MB zero-fill dominates; no matmul structure -> no WMMA.
// CDNA5 path: async global->LDS point streaming + S_WAIT_ASYNCCNT.

#define BEV_GX 512
#define BEV_GY 512
#define BEV_GZ 40
#define BEV_NVOX (BEV_GX * BEV_GY * BEV_GZ)   // 10,485,760 voxels per batch

// The async-to-LDS builtin takes typed int pointers:
//   param0: AS(1) (global / __device__) int*   — source address, per lane
//   param1: AS(3) (LDS / __shared__)    int*   — dest address, per lane
typedef __attribute__((address_space(1))) int* as1_ip;   // global int*
typedef __attribute__((address_space(3))) int* as3_ip;   // LDS int*

// ---------------------------------------------------------------------------
// Pass 1: zero the dense grid with 128-bit stores (global_store_b128).
// Regular (RT) temporal hint on purpose: keep the zeroed lines dirty in the
// 192MB L2 so the immediately-following atomics hit in L2.
// ---------------------------------------------------------------------------
__global__ __launch_bounds__(256) void bev_zero_kernel(float4* __restrict__ out, int n4) {
    int i = blockIdx.x * blockDim.x + threadIdx.x;
    int stride = gridDim.x * blockDim.x;
    const float4 z = make_float4(0.f, 0.f, 0.f, 0.f);
    for (; i < n4; i += stride) {
        out[i] = z;
    }
}

// ---------------------------------------------------------------------------
// Pass 2: one thread per point. Stream the x/y/z planes into LDS with the
// CDNA5 async copy engine (per-lane addressed, ASYNCcnt-tracked), wait, then
// bin and scatter with global_atomic_add_f32 (no return value -> hardware
// no-return atomic, single trip to the L2 atomic units).
// ---------------------------------------------------------------------------
__global__ __launch_bounds__(256) void bev_scatter_kernel(const float* __restrict__ pc,
                                                          float* __restrict__ out,
                                                          int N, int total) {
    const int tid = threadIdx.x;
    const int i   = blockIdx.x * 256 + tid;
    const bool active = (i < total);

    int b = 0, n = 0;
    if (active) { b = i / N; n = i - b * N; }   // per-lane: tile may straddle batches

    float x = 0.f, y = 0.f, z = 0.f;

#if defined(__HIP_DEVICE_COMPILE__) && \
    __has_builtin(__builtin_amdgcn_global_load_async_to_lds_b32) && \
    __has_builtin(__builtin_amdgcn_s_wait_asynccnt)
    // CDNA5 async path: GLOBAL_LOAD_ASYNC_TO_LDS_B32 (ASYNCcnt) + S_WAIT_ASYNCCNT.
    __shared__ float xs[256], ys[256], zs[256];
    if (active) {
        const float* px = pc + (size_t)b * 3u * (size_t)N + (size_t)n;  // x plane
        __builtin_amdgcn_global_load_async_to_lds_b32(
            (as1_ip)(int*)(float*)(px),         (as3_ip)(int*)&xs[tid], 0, 0);
        __builtin_amdgcn_global_load_async_to_lds_b32(
            (as1_ip)(int*)(float*)(px + N),     (as3_ip)(int*)&ys[tid], 0, 0);
        __builtin_amdgcn_global_load_async_to_lds_b32(
            (as1_ip)(int*)(float*)(px + 2 * N), (as3_ip)(int*)&zs[tid], 0, 0);
    }
    __builtin_amdgcn_s_wait_asynccnt(0);  // wave's async LDS writes complete
    __syncthreads();                      // compiler/memory ordering barrier
    x = xs[tid]; y = ys[tid]; z = zs[tid];
#else
    // Fallback: plain coalesced global_load_b32 from the three planes.
    if (active) {
        const float* px = pc + (size_t)b * 3u * (size_t)N + (size_t)n;
        x = px[0];
        y = px[N];
        z = px[2 * N];
    }
#endif

    if (!active) return;

    // Exact match to the reference: f32 subtract, f32 divide by 0.2f,
    // truncation toward zero (C cast == jnp astype(int32) == torch .int()).
    const float VMINX = -51.2f, VMINY = -51.2f, VMINZ = -4.0f, VS = 0.2f;
    int gx = (int)((x - VMINX) / VS);
    int gy = (int)((y - VMINY) / VS);
    int gz = (int)((z - VMINZ) / VS);

    // In-range mask (unsigned compare folds the >=0 and <=max checks).
    if ((unsigned)gx < (unsigned)BEV_GX &&
        (unsigned)gy < (unsigned)BEV_GY &&
        (unsigned)gz < (unsigned)BEV_GZ) {
        size_t lin = (size_t)gx * (BEV_GY * BEV_GZ) + (size_t)gy * BEV_GZ + (size_t)gz;
        size_t o   = ((size_t)b * BEV_NVOX + lin) * 3u;
        atomicAdd(&out[o + 0], x);   // global_atomic_add_f32 (no-return form)
        atomicAdd(&out[o + 1], y);
        atomicAdd(&out[o + 2], z);
    }
}

// ---------------------------------------------------------------------------
// Launch: deterministic two-pass pipeline on `stream`.
// ---------------------------------------------------------------------------
extern "C" void kernel_launch(void* const* d_in, const int* in_sizes, int n_in,
                              void* d_out, int out_size, void* d_ws, size_t ws_size,
                              hipStream_t stream) {
    (void)n_in; (void)d_ws; (void)ws_size;

    const float* pc  = (const float*)d_in[0];   // [B, 3, N] f32
    float*       out = (float*)d_out;           // [B, GX, GY, GZ, 3] f32

    const int B     = out_size / (BEV_NVOX * 3);        // = 2 for this config
    const int total = in_sizes[0] / 3;                  // B * N points
    const int N     = total / (B > 0 ? B : 1);          // points per batch

    // Pass 1: zero grid. out_size = B*512*512*40*3 is divisible by 4.
    const int n4 = out_size / 4;
    const int zthreads = 256, ziters = 8;
    int zblocks = (n4 + zthreads * ziters - 1) / (zthreads * ziters);  // ~7680
    bev_zero_kernel<<<zblocks, zthreads, 0, stream>>>((float4*)out, n4);

    // Pass 2: scatter-add points (1563 blocks of 8 wave32s).
    int sblocks = (total + 255) / 256;
    bev_scatter_kernel<<<sblocks, 256, 0, stream>>>(pc, out, N, total);
}